// GNNModel_60378650247170
// MI455X (gfx1250) — compile-verified
//
#include <hip/hip_runtime.h>

#define DIM     128   // D == H == 128
#define OUT_DIM 64

typedef __attribute__((ext_vector_type(2))) float v2f;
typedef __attribute__((ext_vector_type(8))) float v8f;

// ---------------- degree / normalization ----------------
__global__ void deg_init_kernel(float* __restrict__ deg, int n) {
    int i = blockIdx.x * blockDim.x + threadIdx.x;
    if (i < n) deg[i] = 1.0f;                       // self-loop contribution
}

__global__ void deg_accum_kernel(const int* __restrict__ dst, float* __restrict__ deg, int e) {
    int i = blockIdx.x * blockDim.x + threadIdx.x;
    if (i < e) atomicAdd(&deg[dst[i]], 1.0f);
}

__global__ void deg_finalize_kernel(float* __restrict__ deg, int n) {
    int i = blockIdx.x * blockDim.x + threadIdx.x;
    if (i < n) deg[i] = rsqrtf(deg[i]);             // deg >= 1 always (self-loop)
}

// ---------------- fp32 WMMA GEMM: hp = (A @ W) * dinv[row]; acc = hp ----------------
// A: [N,128], W: [128,128]. Block = 256 threads = 8 waves; tile = 16 rows x 128 cols.
// acc may alias A (in-place): block stages its own 16 rows to LDS first, writes only those rows.
__global__ __launch_bounds__(256) void gemm128_scale_kernel(
    const float* __restrict__ A, const float* __restrict__ W,
    const float* __restrict__ dinv,
    float* __restrict__ hp, float* __restrict__ acc)
{
    __shared__ __align__(16) float As[16 * DIM];
    const int row0 = blockIdx.x * 16;
    const int t = threadIdx.x;
    {
        const float4* s4 = (const float4*)(A + (size_t)row0 * DIM);
        float4* d4 = (float4*)As;
        d4[t]       = s4[t];
        d4[t + 256] = s4[t + 256];
    }
    __syncthreads();

    const int lane = t & 31;
    const int wave = t >> 5;
    const int m  = lane & 15;                // A row within tile
    const int kb = (lane >> 4) << 1;         // K sub-offset: lanes 0-15 -> {0,1}, 16-31 -> {2,3}
    const int n  = (lane & 15) + (wave << 4); // output column 0..127

    v8f c = {};
#pragma unroll
    for (int k0 = 0; k0 < DIM; k0 += 4) {
        v2f a = *(const v2f*)(&As[m * DIM + k0 + kb]);
        v2f b;
        b.x = W[(size_t)(k0 + kb)     * DIM + n];
        b.y = W[(size_t)(k0 + kb + 1) * DIM + n];
        c = __builtin_amdgcn_wmma_f32_16x16x4_f32(false, a, false, b, (short)0, c, false, false);
    }

    const int rb = (lane < 16) ? 0 : 8;      // C/D layout: VGPR g -> row g (lanes 0-15) / g+8 (16-31)
#pragma unroll
    for (int g = 0; g < 8; ++g) {
        const int r = row0 + rb + g;
        const float v = c[g] * dinv[r];
        hp [(size_t)r * DIM + n] = v;
        acc[(size_t)r * DIM + n] = v;        // self-loop init of accumulator
    }
}

// ---------------- edge scatter: acc[dst] += hp[src], one wave per edge ----------------
__global__ __launch_bounds__(256) void scatter_kernel(
    const float* __restrict__ hp, float* __restrict__ acc,
    const int* __restrict__ src, const int* __restrict__ dst, int e)
{
    const int wid  = (int)((blockIdx.x * blockDim.x + threadIdx.x) >> 5);
    const int lane = threadIdx.x & 31;
    if (wid >= e) return;
    const int s = src[wid];
    const int d = dst[wid];
    const float4 v = ((const float4*)(hp + (size_t)s * DIM))[lane];   // coalesced 512B/wave
    float* a = acc + (size_t)d * DIM + lane * 4;
    atomicAdd(a + 0, v.x);
    atomicAdd(a + 1, v.y);
    atomicAdd(a + 2, v.z);
    atomicAdd(a + 3, v.w);
}

// ---------------- out = relu(acc * dinv[row] + b) ----------------
__global__ void finish_kernel(const float* __restrict__ acc, const float* __restrict__ dinv,
                              const float* __restrict__ bias, float* __restrict__ out, int total)
{
    const int i = blockIdx.x * blockDim.x + threadIdx.x;
    if (i >= total) return;
    const int r = i >> 7;
    const float v = fmaf(acc[i], dinv[r], bias[i & (DIM - 1)]);
    out[i] = v > 0.0f ? v : 0.0f;
}

// ---------------- final head: out = A @ Wf + bf  (Wf: [128,64]) ----------------
__global__ __launch_bounds__(128) void gemm_final_kernel(
    const float* __restrict__ A, const float* __restrict__ W,
    const float* __restrict__ bias, float* __restrict__ out)
{
    __shared__ __align__(16) float As[16 * DIM];
    const int row0 = blockIdx.x * 16;
    const int t = threadIdx.x;
    {
        const float4* s4 = (const float4*)(A + (size_t)row0 * DIM);
        float4* d4 = (float4*)As;
#pragma unroll
        for (int i = 0; i < 4; ++i) d4[t + i * 128] = s4[t + i * 128];
    }
    __syncthreads();

    const int lane = t & 31;
    const int wave = t >> 5;                  // 0..3 -> columns 0..63
    const int m  = lane & 15;
    const int kb = (lane >> 4) << 1;
    const int n  = (lane & 15) + (wave << 4);

    v8f c = {};
#pragma unroll
    for (int k0 = 0; k0 < DIM; k0 += 4) {
        v2f a = *(const v2f*)(&As[m * DIM + k0 + kb]);
        v2f b;
        b.x = W[(size_t)(k0 + kb)     * OUT_DIM + n];
        b.y = W[(size_t)(k0 + kb + 1) * OUT_DIM + n];
        c = __builtin_amdgcn_wmma_f32_16x16x4_f32(false, a, false, b, (short)0, c, false, false);
    }

    const float bv = bias[n];
    const int rb = (lane < 16) ? 0 : 8;
#pragma unroll
    for (int g = 0; g < 8; ++g) {
        const int r = row0 + rb + g;
        out[(size_t)r * OUT_DIM + n] = c[g] + bv;
    }
}

extern "C" void kernel_launch(void* const* d_in, const int* in_sizes, int n_in,
                              void* d_out, int out_size, void* d_ws, size_t ws_size,
                              hipStream_t stream)
{
    const float* x  = (const float*)d_in[0];
    const float* W1 = (const float*)d_in[1];
    const float* b1 = (const float*)d_in[2];
    const float* W2 = (const float*)d_in[3];
    const float* b2 = (const float*)d_in[4];
    const float* Wf = (const float*)d_in[5];
    const float* bf = (const float*)d_in[6];
    const int*   ei = (const int*)d_in[7];

    const int N = in_sizes[0] / DIM;     // 100000 (divisible by 16)
    const int E = in_sizes[7] / 2;       // 1600000
    const int* src = ei;
    const int* dst = ei + E;

    // workspace: dinv [N] (padded) + two N x 128 fp32 buffers (~103 MB total)
    float* dinv = (float*)d_ws;
    float* buf0 = dinv + (((size_t)N + 255) & ~(size_t)255);
    float* buf1 = buf0 + (size_t)N * DIM;

    const int total  = N * DIM;
    const int nTiles = N / 16;

    // symmetric normalization coefficients (shared by both layers)
    deg_init_kernel    <<<(N + 255) / 256, 256, 0, stream>>>(dinv, N);
    deg_accum_kernel   <<<(E + 255) / 256, 256, 0, stream>>>(dst, dinv, E);
    deg_finalize_kernel<<<(N + 255) / 256, 256, 0, stream>>>(dinv, N);

    // layer 1: hp=buf0, acc=buf1
    gemm128_scale_kernel<<<nTiles, 256, 0, stream>>>(x, W1, dinv, buf0, buf1);
    scatter_kernel<<<(E + 7) / 8, 256, 0, stream>>>(buf0, buf1, src, dst, E);
    finish_kernel <<<(total + 255) / 256, 256, 0, stream>>>(buf1, dinv, b1, buf0, total);

    // layer 2: hp=buf1, acc=buf0 (acc in-place over GEMM input; safe per-row)
    gemm128_scale_kernel<<<nTiles, 256, 0, stream>>>(buf0, W2, dinv, buf1, buf0);
    scatter_kernel<<<(E + 7) / 8, 256, 0, stream>>>(buf1, buf0, src, dst, E);
    finish_kernel <<<(total + 255) / 256, 256, 0, stream>>>(buf0, dinv, b2, buf1, total);

    // head
    gemm_final_kernel<<<nTiles, 128, 0, stream>>>(buf1, Wf, bf, (float*)d_out);
}